// Net_83434034692739
// MI455X (gfx1250) — compile-verified
//
#include <hip/hip_runtime.h>
#include <math.h>

typedef float v2f __attribute__((ext_vector_type(2)));
typedef float v8f __attribute__((ext_vector_type(8)));

// ---------------------------------------------------------------------------
// Elementwise fill
// ---------------------------------------------------------------------------
__global__ void fill_kernel(float* __restrict__ p, float v, int n) {
  int i = blockIdx.x * blockDim.x + threadIdx.x;
  if (i < n) p[i] = v;
}

// ---------------------------------------------------------------------------
// deg[col] += 1 for every real edge (deg pre-initialized to 1.0 for self loop)
// edge_index is int64 [2, E]
// ---------------------------------------------------------------------------
__global__ void degree_kernel(const long long* __restrict__ ei,
                              float* __restrict__ deg, int E) {
  int e = blockIdx.x * blockDim.x + threadIdx.x;
  if (e < E) atomicAdd(&deg[(int)ei[(size_t)E + e]], 1.0f);
}

__global__ void dinv_kernel(const float* __restrict__ deg,
                            float* __restrict__ dinv, int n) {
  int i = blockIdx.x * blockDim.x + threadIdx.x;
  if (i < n) {
    float d = deg[i];
    dinv[i] = d > 0.0f ? rsqrtf(d) : 0.0f;
  }
}

// ---------------------------------------------------------------------------
// C[M x N] = A[M x K] @ B[K x N]  via V_WMMA_F32_16X16X4_F32 (exact f32).
// One wave computes one 16x16 output tile. M,N multiples of 16; K multiple of 4.
//
// Per the CDNA5 ISA VGPR layouts (CDNA5 ISA):
//   lane l: m = l&15, hi = l>>4
//   A (16x4 f32):  a[0]=A[m][k+2*hi],   a[1]=A[m][k+2*hi+1]   (contiguous pair)
//   B (4x16 f32):  b[0]=B[k+2*hi][n],   b[1]=B[k+2*hi+1][n],  n = l&15
//   C (16x16 f32): acc[v] = C[v + 8*hi][n]
// ---------------------------------------------------------------------------
__global__ void gemm_f32_wmma(const float* __restrict__ A,
                              const float* __restrict__ B,
                              float* __restrict__ C,
                              int M, int N, int K) {
  const int wavesPerBlock = blockDim.x >> 5;
  const int tile = blockIdx.x * wavesPerBlock + (threadIdx.x >> 5);
  const int ntiles = N >> 4;
  const int mtiles = M >> 4;
  if (tile >= mtiles * ntiles) return;            // wave-uniform: EXEC stays all-1s
  const int mt = tile / ntiles;
  const int nt = tile - mt * ntiles;
  const int lane = threadIdx.x & 31;
  const int lm = lane & 15;                       // A row / B,C column within tile
  const int hi = lane >> 4;

  const float* __restrict__ arow = A + (size_t)(mt * 16 + lm) * K + 2 * hi;
  const float* __restrict__ bcol = B + (size_t)(nt * 16 + lm);

  v8f acc = {};
  for (int k = 0; k < K; k += 4) {
    v2f a = *(const v2f*)(arow + k);
    v2f b;
    b[0] = bcol[(size_t)(k + 2 * hi + 0) * N];
    b[1] = bcol[(size_t)(k + 2 * hi + 1) * N];
    acc = __builtin_amdgcn_wmma_f32_16x16x4_f32(
        /*neg_a=*/false, a, /*neg_b=*/false, b,
        /*c_mod=*/(short)0, acc, /*reuse_a=*/false, /*reuse_b=*/false);
  }

  float* __restrict__ crow = C + (size_t)(mt * 16 + 8 * hi) * N + nt * 16 + lm;
#pragma unroll
  for (int v = 0; v < 8; ++v) crow[(size_t)v * N] = acc[v];
}

// ---------------------------------------------------------------------------
// Layer-1 aggregation (F = 128): one wave per edge, one float4 per lane.
// agg[col] += h[row] * (dinv[row]*dinv[col]).  Edges e in [E, E+Nn) are the
// self loops (row = col = e - E).
// ---------------------------------------------------------------------------
__global__ void scatter_f128(const long long* __restrict__ ei,
                             const float* __restrict__ dinv,
                             const float* __restrict__ h,
                             float* __restrict__ agg,
                             int E, int Nn) {
  const int edgesPerBlock = blockDim.x >> 5;
  const long long edge =
      (long long)blockIdx.x * edgesPerBlock + (threadIdx.x >> 5);
  const long long total = (long long)E + Nn;
  if (edge >= total) return;
  const int lane = threadIdx.x & 31;

  int r, c;
  if (edge < E) {
    r = (int)ei[edge];
    c = (int)ei[(size_t)E + edge];
  } else {
    r = c = (int)(edge - E);
  }
  const float nrm = dinv[r] * dinv[c];

  const float4 v = *(const float4*)(h + (size_t)r * 128 + lane * 4);
  float* dst = agg + (size_t)c * 128 + lane * 4;
  atomicAdd(dst + 0, v.x * nrm);
  atomicAdd(dst + 1, v.y * nrm);
  atomicAdd(dst + 2, v.z * nrm);
  atomicAdd(dst + 3, v.w * nrm);
}

// a[i] = relu(a[i] + bias[i % F]), F power of two
__global__ void bias_relu_kernel(float* __restrict__ a,
                                 const float* __restrict__ b, int n, int Fmask) {
  int i = blockIdx.x * blockDim.x + threadIdx.x;
  if (i < n) a[i] = fmaxf(a[i] + b[i & Fmask], 0.0f);
}

// ---------------------------------------------------------------------------
// Layer-2 aggregation (F = 16): one thread per (edge, feature).
// ---------------------------------------------------------------------------
__global__ void scatter_f16(const long long* __restrict__ ei,
                            const float* __restrict__ dinv,
                            const float* __restrict__ h,
                            float* __restrict__ agg,
                            int E, int Nn) {
  const long long idx = (long long)blockIdx.x * blockDim.x + threadIdx.x;
  const long long total = ((long long)E + Nn) * 16;
  if (idx >= total) return;
  const long long edge = idx >> 4;
  const int f = (int)(idx & 15);

  int r, c;
  if (edge < E) {
    r = (int)ei[edge];
    c = (int)ei[(size_t)E + edge];
  } else {
    r = c = (int)(edge - E);
  }
  const float nrm = dinv[r] * dinv[c];
  atomicAdd(agg + (size_t)c * 16 + f, h[(size_t)r * 16 + f] * nrm);
}

// out[i,:] = log_softmax(agg[i,:] + b2), row width 16, one thread per row
__global__ void logsoftmax16_kernel(const float* __restrict__ agg,
                                    const float* __restrict__ b,
                                    float* __restrict__ out, int n) {
  int i = blockIdx.x * blockDim.x + threadIdx.x;
  if (i >= n) return;
  float vals[16];
  float mx = -1e30f;
#pragma unroll
  for (int f = 0; f < 16; ++f) {
    float v = agg[(size_t)i * 16 + f] + b[f];
    vals[f] = v;
    mx = fmaxf(mx, v);
  }
  float s = 0.0f;
#pragma unroll
  for (int f = 0; f < 16; ++f) s += expf(vals[f] - mx);
  const float lse = mx + logf(s);
#pragma unroll
  for (int f = 0; f < 16; ++f) out[(size_t)i * 16 + f] = vals[f] - lse;
}

// ---------------------------------------------------------------------------
extern "C" void kernel_launch(void* const* d_in, const int* in_sizes, int n_in,
                              void* d_out, int out_size, void* d_ws,
                              size_t ws_size, hipStream_t stream) {
  const float*     x  = (const float*)d_in[0];      // [Nn,128] f32
  const long long* ei = (const long long*)d_in[1];  // [2,E]    int64
  const float*     W1 = (const float*)d_in[2];      // [128,128]
  const float*     b1 = (const float*)d_in[3];      // [128]
  const float*     W2 = (const float*)d_in[4];      // [128,16]
  const float*     b2 = (const float*)d_in[5];      // [16]
  float*           out = (float*)d_out;             // [Nn,16]

  const int Nn = in_sizes[0] / 128;  // 50000
  const int E  = in_sizes[1] / 2;    // 800000

  // workspace layout (floats)
  float* ws   = (float*)d_ws;
  float* deg  = ws;                              // Nn
  float* dinv = deg + Nn;                        // Nn
  float* h1   = dinv + Nn;                       // Nn*128
  float* agg1 = h1 + (size_t)Nn * 128;           // Nn*128
  float* h2   = agg1 + (size_t)Nn * 128;         // Nn*16
  float* agg2 = h2 + (size_t)Nn * 16;            // Nn*16

  // init: deg = 1.0 (self loop), agg buffers = 0
  fill_kernel<<<(Nn + 255) / 256, 256, 0, stream>>>(deg, 1.0f, Nn);
  fill_kernel<<<(Nn * 128 + 255) / 256, 256, 0, stream>>>(agg1, 0.0f, Nn * 128);
  fill_kernel<<<(Nn * 16 + 255) / 256, 256, 0, stream>>>(agg2, 0.0f, Nn * 16);

  // degree + symmetric normalization
  degree_kernel<<<(E + 255) / 256, 256, 0, stream>>>(ei, deg, E);
  dinv_kernel<<<(Nn + 255) / 256, 256, 0, stream>>>(deg, dinv, Nn);

  // h1 = x @ W1   (M=Nn, K=128, N=128)
  {
    const int tiles = (Nn / 16) * (128 / 16);
    const int wpb = 4;
    gemm_f32_wmma<<<(tiles + wpb - 1) / wpb, wpb * 32, 0, stream>>>(
        x, W1, h1, Nn, 128, 128);
  }

  // agg1 = scatter(h1, norm); then + b1, relu
  {
    const long long total = (long long)E + Nn;
    const int epb = 8;  // edges per 256-thread block
    scatter_f128<<<(int)((total + epb - 1) / epb), epb * 32, 0, stream>>>(
        ei, dinv, h1, agg1, E, Nn);
  }
  bias_relu_kernel<<<(Nn * 128 + 255) / 256, 256, 0, stream>>>(agg1, b1,
                                                               Nn * 128, 127);

  // h2 = relu(agg1) @ W2   (M=Nn, K=128, N=16)
  {
    const int tiles = (Nn / 16) * (16 / 16);
    const int wpb = 4;
    gemm_f32_wmma<<<(tiles + wpb - 1) / wpb, wpb * 32, 0, stream>>>(
        agg1, W2, h2, Nn, 16, 128);
  }

  // agg2 = scatter(h2, norm)
  {
    const long long total = ((long long)E + Nn) * 16;
    scatter_f16<<<(int)((total + 255) / 256), 256, 0, stream>>>(ei, dinv, h2,
                                                                agg2, E, Nn);
  }

  // out = log_softmax(agg2 + b2)
  logsoftmax16_kernel<<<(Nn + 255) / 256, 256, 0, stream>>>(agg2, b2, out, Nn);
}